// BatchRecallLoss_51737176047663
// MI455X (gfx1250) — compile-verified
//
#include <hip/hip_runtime.h>
#include <stdint.h>

// BatchRecallLoss on MI455X (gfx1250).
// Bandwidth-bound single pass over 80 MB (floor ~3.4us @ 23.3 TB/s).
// CDNA5 async global->LDS DMA (ASYNCcnt path), double-buffered:
//   global_load_async_to_lds_b128 + s_wait_asynccnt.
// All shape constants are compile-time: no runtime division, fully
// unrolled tile pipeline, wave32 shuffle reductions.

#define NN     16                  // batch
#define CC     4                   // classes
#define PP     262144              // spatial size
#define TPB    256                 // threads per block (8 wave32)
#define VEC    4                   // float4 per thread per tile
#define TILE   (TPB * VEC)         // 1024 p-elements per tile
#define CHUNK  8192                // p-elements per block
#define NTILES (CHUNK / TILE)      // 8 tiles -> deep async pipeline
#define BLKX   (PP / CHUNK)        // 32 blocks along P
#define NWAVE  (TPB / 32)          // 8 waves

// ---- CDNA5 async global->LDS copy (tracked by ASYNCcnt) --------------------
__device__ __forceinline__ void async_b128(uint32_t lds_addr, const void* gaddr) {
    asm volatile("global_load_async_to_lds_b128 %0, %1, off"
                 :: "v"(lds_addr), "v"((uint64_t)(uintptr_t)gaddr)
                 : "memory");
}
// After issuing the next tile's 5 async ops, <=5 outstanding means the
// previous tile's 5 ops completed (async loads complete in order).
__device__ __forceinline__ void wait_async_le5() {
    asm volatile("s_wait_asynccnt 0x5" ::: "memory");
}
__device__ __forceinline__ void wait_async_0() {
    asm volatile("s_wait_asynccnt 0x0" ::: "memory");
}

__global__ __launch_bounds__(TPB)
void brl_main(const float* __restrict__ pred,   // (N, C, P)
              const int*   __restrict__ tgt,    // (N, P)
              float*       __restrict__ part)   // (N*BLKX, 12) partials
{
    __shared__ float4 sPred[2][CC][TPB];   // 32 KB staging
    __shared__ int4   sTgt [2][TPB];       //  8 KB staging
    __shared__ float  sWave[NWAVE][12];    // cross-wave reduction

    const int tid = threadIdx.x;
    const int n   = blockIdx.y;
    const long pbase = (long)blockIdx.x * CHUNK + tid * VEC;

    const float* pn = pred + (size_t)n * CC * PP;
    const int*   tn = tgt  + (size_t)n * PP;

    auto issue = [&](int j, int buf) {
        const long p = pbase + (long)j * TILE;
#pragma unroll
        for (int c = 0; c < CC; ++c)
            async_b128((uint32_t)(uintptr_t)&sPred[buf][c][tid],
                       pn + (size_t)c * PP + p);
        async_b128((uint32_t)(uintptr_t)&sTgt[buf][tid], tn + p);
    };

    float nlla[CC] = {0.f, 0.f, 0.f, 0.f};
    int   tpa [CC] = {0, 0, 0, 0};
    int   posa[CC] = {0, 0, 0, 0};

    issue(0, 0);                                // prologue
#pragma unroll
    for (int j = 0; j < NTILES; ++j) {
        const int buf = j & 1;
        if (j + 1 < NTILES) { issue(j + 1, buf ^ 1); wait_async_le5(); }
        else                { wait_async_0(); }

        // Each lane consumes exactly the LDS region its own async ops wrote:
        // no workgroup barrier needed in the streaming loop.
        float4 xc[CC];
#pragma unroll
        for (int c = 0; c < CC; ++c) xc[c] = sPred[buf][c][tid];
        const int4 t4 = sTgt[buf][tid];
        const int te[4] = {t4.x, t4.y, t4.z, t4.w};

#pragma unroll
        for (int e = 0; e < VEC; ++e) {
            float x[CC];
#pragma unroll
            for (int c = 0; c < CC; ++c)
                x[c] = reinterpret_cast<const float*>(&xc[c])[e];
            const int t = te[e];

            // argmax (first occurrence, like jnp.argmax)
            float m = x[0]; int am = 0;
#pragma unroll
            for (int c = 1; c < CC; ++c) if (x[c] > m) { m = x[c]; am = c; }

            // log-softmax NLL
            float se = 0.f;
#pragma unroll
            for (int c = 0; c < CC; ++c) se += __expf(x[c] - m);
            float xt = x[0];
#pragma unroll
            for (int c = 1; c < CC; ++c) xt = (t == c) ? x[c] : xt;
            const float nll = __logf(se) + m - xt;

#pragma unroll
            for (int c = 0; c < CC; ++c) {
                const int is = (t == c);
                posa[c] += is;
                tpa [c] += is & (am == c);
                nlla[c] += is ? nll : 0.f;
            }
        }
    }

    // ---- block reduction: wave32 butterflies, then one LDS pass ------------
    float v[12];
#pragma unroll
    for (int c = 0; c < CC; ++c) {
        v[c]     = (float)tpa[c];
        v[4 + c] = (float)posa[c];
        v[8 + c] = nlla[c];
    }
#pragma unroll
    for (int off = 16; off > 0; off >>= 1) {
#pragma unroll
        for (int k = 0; k < 12; ++k) v[k] += __shfl_xor(v[k], off, 32);
    }
    const int wid = tid >> 5, lane = tid & 31;
    if (lane == 0) {
#pragma unroll
        for (int k = 0; k < 12; ++k) sWave[wid][k] = v[k];
    }
    __syncthreads();
    if (tid < 12) {
        float s = 0.f;
#pragma unroll
        for (int w = 0; w < NWAVE; ++w) s += sWave[w][tid];   // fixed order
        part[(size_t)(n * BLKX + blockIdx.x) * 12 + tid] = s;
    }
}

__global__ __launch_bounds__(64)
void brl_final(const float* __restrict__ part, float* __restrict__ out)
{
    __shared__ float sTp[NN][CC], sPos[NN][CC], sNll[NN][CC];
    const int tid = threadIdx.x;
    if (tid < NN * CC) {
        const int n = tid / CC, c = tid % CC;
        float tp = 0.f, pos = 0.f, nll = 0.f;
        for (int b = 0; b < BLKX; ++b) {             // fixed order: deterministic
            const float* o = part + (size_t)(n * BLKX + b) * 12;
            tp += o[c]; pos += o[4 + c]; nll += o[8 + c];
        }
        sTp[n][c] = tp; sPos[n][c] = pos; sNll[n][c] = nll;
    }
    __syncthreads();
    if (tid == 0) {
        float num = 0.f, den = 0.f;
        for (int c = 0; c < CC; ++c) {
            float rsum = 0.f, posT = 0.f, nllT = 0.f;
            for (int n = 0; n < NN; ++n) {
                const float pos = sPos[n][c];
                rsum += (pos > 0.f) ? (sTp[n][c] / pos) : 1.0f;
                posT += pos;
                nllT += sNll[n][c];
            }
            const float w = 1.0f - rsum / (float)NN;  // weight[c]
            num += w * nllT;                           // sum(w * nll)
            den += w * posT;                           // sum(w)
        }
        out[0] = num / den;
    }
}

extern "C" void kernel_launch(void* const* d_in, const int* in_sizes, int n_in,
                              void* d_out, int out_size, void* d_ws, size_t ws_size,
                              hipStream_t stream) {
    const float* pred = (const float*)d_in[0];   // (16, 4, 262144) f32
    const int*   tgt  = (const int*)d_in[1];     // (16, 262144) i32
    float* part = (float*)d_ws;                  // NN*BLKX*12 floats = 24 KB

    dim3 grid(BLKX, NN);
    brl_main<<<grid, TPB, 0, stream>>>(pred, tgt, part);
    brl_final<<<1, 64, 0, stream>>>(part, (float*)d_out);
}